// AggregationNonCupy_56796647522327
// MI455X (gfx1250) — compile-verified
//
#include <hip/hip_runtime.h>

// ---------------------------------------------------------------------------
// Aggregation (SAN-style):  out[n, g*32+c, y, x] =
//     sum_{i,j in 0..2} weight[n, c, i*3+j, y*64+x] * in[n, g*32+c, y+i-1, x+j-1]
// Memory-bound (~1.7 FLOP/B): streaming kernel with double-buffered async
// global->LDS staging (ASYNCcnt), B128 accesses, NT stores, weight reuse
// across the 16 groups, and weights hoisted to VGPRs.
// ---------------------------------------------------------------------------

typedef float v4f __attribute__((ext_vector_type(4)));
typedef int   v4i __attribute__((vector_size(16)));   // matches builtin pointee

#define GAS __attribute__((address_space(1)))
#define LAS __attribute__((address_space(3)))

#define N_      16
#define CX      512
#define HW      64
#define CW      32
#define GROUPS  16
#define OHW     4096          // 64*64 pixels per plane
#define ROWS    16            // output rows per block
#define PIX     (ROWS * 64)   // 1024 pixels per block tile
#define XROWS   (ROWS + 2)    // 18 staged input rows (with y halo)
#define XSTR    72            // LDS row stride in floats: 4 pad | 64 data | 4 pad
#define XTILE   (XROWS * XSTR)   // 1296 floats per input buffer
#define WLDS    (9 * PIX)        // 9216 floats of weights

__device__ __forceinline__ void stage_b128(const float* gsrc, float* ldst) {
#if __has_builtin(__builtin_amdgcn_global_load_async_to_lds_b128)
  // async DMA: global -> LDS, 128 bits per lane, tracked by ASYNCcnt
  __builtin_amdgcn_global_load_async_to_lds_b128(
      (GAS v4i*)(float*)gsrc, (LAS v4i*)ldst, 0, 0);
#else
  *(LAS v4f*)ldst = *(const GAS v4f*)gsrc;   // fallback: B128 load + DS store
#endif
}

template <int CNT>
__device__ __forceinline__ void wait_async() {
#if __has_builtin(__builtin_amdgcn_s_wait_asynccnt)
  __builtin_amdgcn_s_wait_asynccnt(CNT);
#else
  asm volatile("s_wait_asynccnt %0" ::"i"(CNT) : "memory");
#endif
}

// Stage the 18x64-float tile (288 B128 transfers) with EXACTLY 2 async ops per
// wave: wave w covers transfers [36w, 36w+36) as one full-EXEC op + one 4-lane
// op. Deterministic per-wave ASYNCcnt lets us use s_wait_asynccnt immediates.
__device__ __forceinline__ void stage_input(const float* plane_ptr, int y0,
                                            int wv, int ln, float* buf) {
  {
    const int idx = 36 * wv + ln;                       // 0..283
    const int lr = idx >> 4, cf = idx & 15;
    int gy = y0 - 1 + lr;
    gy = gy < 0 ? 0 : (gy > 63 ? 63 : gy);              // clamp; fixed in compute
    stage_b128(plane_ptr + gy * 64 + cf * 4, buf + lr * XSTR + 4 + cf * 4);
  }
  if (ln < 4) {
    const int idx = 36 * wv + 32 + ln;                  // 32..287
    const int lr = idx >> 4, cf = idx & 15;
    int gy = y0 - 1 + lr;
    gy = gy < 0 ? 0 : (gy > 63 ? 63 : gy);
    stage_b128(plane_ptr + gy * 64 + cf * 4, buf + lr * XSTR + 4 + cf * 4);
  }
}

__global__ __launch_bounds__(256) void aggregation_kernel(
    const float* __restrict__ in, const float* __restrict__ wgt,
    float* __restrict__ out) {
  __shared__ float smem[WLDS + 2 * XTILE];
  float* wl  = smem;            // [9][1024] weight staging tile
  float* xb0 = smem + WLDS;     // input tile buffer 0
  float* xb1 = xb0 + XTILE;     // input tile buffer 1

  const int t  = threadIdx.x;
  const int wv = t >> 5, ln = t & 31;
  const int y0 = blockIdx.x * ROWS;   // first output row of this tile
  const int c  = blockIdx.y;          // channel-within-group (weight channel)
  const int n  = blockIdx.z;
  const int p0 = y0 * 64;

  // ---- stage the 9 x 1024 weight tile once (9 async ops per wave) ----
  const size_t wbase = (((size_t)n * CW + c) * 9) * OHW + p0;
#pragma unroll
  for (int k = 0; k < 9; ++k)
    stage_b128(wgt + wbase + (size_t)k * OHW + t * 4, wl + k * PIX + t * 4);

  // ---- zero the pad columns (0..3, 68..71) of both buffers, once ----
  for (int idx = t; idx < 2 * XROWS * 8; idx += 256) {
    const int b  = idx / (XROWS * 8);
    const int rr = (idx >> 3) % XROWS;
    const int cc = idx & 7;
    (b ? xb1 : xb0)[rr * XSTR + (cc < 4 ? cc : 64 + cc)] = 0.0f;
  }

  // ---- prefetch group 0 input tile (2 async ops per wave) ----
  const size_t plane0 = ((size_t)n * CX + c) * OHW;
  stage_input(in + plane0, y0, wv, ln, xb0);

  // <=2 outstanding can only be the xb0 stage (in-order) => weights landed
  wait_async<2>();
  __syncthreads();

  // ---- hoist this thread's 36 weights into VGPRs (loop-invariant) ----
  v4f wr[9];
#pragma unroll
  for (int k = 0; k < 9; ++k)
    wr[k] = *(const v4f*)(wl + k * PIX + t * 4);

  const int r   = t >> 4;          // local output row 0..15
  const int x0  = (t & 15) << 2;   // output col 0..60 (4 pixels per thread)
  const int gyb = y0 + r - 1;      // input row for i=0

  for (int g = 0; g < GROUPS; ++g) {
    float* cur = (g & 1) ? xb1 : xb0;
    const size_t plane = ((size_t)n * CX + (size_t)g * CW + c) * OHW;

    if (g + 1 < GROUPS) {
      // prefetch next group's tile into the other buffer, then wait for all
      // ops EXCEPT the 2 just issued (async loads complete in order)
      stage_input(in + plane + (size_t)CW * OHW, y0, wv, ln,
                  (g & 1) ? xb0 : xb1);
      wait_async<2>();
    } else {
      wait_async<0>();
    }
    __syncthreads();

    v4f acc = {0.f, 0.f, 0.f, 0.f};
#pragma unroll
    for (int i = 0; i < 3; ++i) {
      const bool vld = (unsigned)(gyb + i) < 64u;      // y-halo rows are zero
      const float* row = cur + (r + i) * XSTR + 3 + x0;
#pragma unroll
      for (int j = 0; j < 3; ++j) {
#pragma unroll
        for (int u = 0; u < 4; ++u) {
          const float xv = vld ? row[u + j] : 0.0f;
          acc[u] = fmaf(wr[i * 3 + j][u], xv, acc[u]);
        }
      }
    }

    // output is never re-read: non-temporal 128-bit store
    __builtin_nontemporal_store(acc, (v4f*)(out + plane + p0 + t * 4));

    __syncthreads();   // next iteration's DMA overwrites the buffer just read
  }
}

extern "C" void kernel_launch(void* const* d_in, const int* in_sizes, int n_in,
                              void* d_out, int out_size, void* d_ws, size_t ws_size,
                              hipStream_t stream) {
  const float* in  = (const float*)d_in[0];   // (16, 512, 64, 64) f32
  const float* wgt = (const float*)d_in[1];   // (16, 32, 9, 4096) f32
  float* out = (float*)d_out;                 // (16, 512, 64, 64) f32

  dim3 grid(HW / ROWS, CW, N_);               // (4, 32, 16) blocks
  aggregation_kernel<<<grid, 256, 0, stream>>>(in, wgt, out);
}